// GTCTrainer_64458869178865
// MI455X (gfx1250) — compile-verified
//
#include <hip/hip_runtime.h>
#include <math.h>

typedef float v2f __attribute__((ext_vector_type(2)));
typedef float v8f __attribute__((ext_vector_type(8)));

#define NN 100000   // nodes
#define EE 1200000  // edges
#define SS 8192     // segments

static __device__ __forceinline__ v8f wmma4(v2f a, v2f b, v8f c) {
  // D = A(16x4 f32) * B(4x16 f32) + C(16x16 f32)
  return __builtin_amdgcn_wmma_f32_16x16x4_f32(false, a, false, b, (short)0, c,
                                               false, false);
}

static __device__ __forceinline__ void wait_asynccnt0() {
#if __has_builtin(__builtin_amdgcn_s_wait_asynccnt)
  __builtin_amdgcn_s_wait_asynccnt(0);
#else
  asm volatile("s_wait_asynccnt 0x0" ::: "memory");
#endif
}

// Async copy of 16 bytes global -> LDS (gfx1250 GLOBAL_LOAD_ASYNC_TO_LDS_B128,
// tracked by ASYNCcnt; LDS address is the wave-relative low-32 of flat addr).
static __device__ __forceinline__ void async_g2l_b128(const float* g,
                                                      const float* lds) {
  unsigned l32 = (unsigned)(unsigned long long)(const void*)lds;
  unsigned long long ga = (unsigned long long)(const void*)g;
  asm volatile("global_load_async_to_lds_b128 %0, %1, off" ::"v"(l32), "v"(ga)
               : "memory");
}

// ------------- h_self = relu([x | cos(t*freq+phase)] @ Wt(128x64) + bt) -----
__global__ __launch_bounds__(128) void k_hself(
    const float* __restrict__ x, const float* __restrict__ ts,
    const float* __restrict__ freq, const float* __restrict__ phase,
    const float* __restrict__ Wt, const float* __restrict__ bt,
    float* __restrict__ hself) {
  __shared__ __align__(16) float sA[64 * 132];  // 64 x 128, stride 132
  const int tid = threadIdx.x;
  const int row0 = blockIdx.x * 64;

  // vectorized fill: 64 rows x 32 float4
  for (int i = tid; i < 64 * 32; i += 128) {
    int r = i >> 5, q = i & 31;
    int gr = row0 + r;
    if (gr >= NN) gr = NN - 1;
    float4 v;
    if (q < 16) {
      v = reinterpret_cast<const float4*>(x)[gr * 16 + q];
    } else {
      int j = q - 16;
      float t = ts[gr];
      float4 f = reinterpret_cast<const float4*>(freq)[j];
      float4 p = reinterpret_cast<const float4*>(phase)[j];
      v.x = cosf(t * f.x + p.x);
      v.y = cosf(t * f.y + p.y);
      v.z = cosf(t * f.z + p.z);
      v.w = cosf(t * f.w + p.w);
    }
    *reinterpret_cast<float4*>(&sA[r * 132 + q * 4]) = v;
  }
  __syncthreads();

  const int lane = tid & 31;
  const int wave = tid >> 5;
  const int m = lane & 15;
  const int koff = (lane >> 4) << 1;  // upper half-wave: K+2,K+3
  const int rsel = (lane >> 4) << 3;  // upper half-wave: rows M+8..M+15

  v8f acc[4];
#pragma unroll
  for (int t = 0; t < 4; ++t) {
    float b = bt[t * 16 + m];
#pragma unroll
    for (int i = 0; i < 8; ++i) acc[t][i] = b;
  }

  const float* Arow = &sA[(wave * 16 + m) * 132];
#pragma unroll 4
  for (int k = 0; k < 128; k += 4) {
    v2f a;
    a.x = Arow[k + koff];
    a.y = Arow[k + koff + 1];
#pragma unroll
    for (int t = 0; t < 4; ++t) {
      v2f b;
      b.x = Wt[(k + koff) * 64 + t * 16 + m];
      b.y = Wt[(k + koff + 1) * 64 + t * 16 + m];
      acc[t] = wmma4(a, b, acc[t]);
    }
  }

  const int gr0 = row0 + wave * 16 + rsel;
  if (row0 + 64 <= NN) {  // uniform fast path: no per-element guards
#pragma unroll
    for (int t = 0; t < 4; ++t) {
      int c = t * 16 + m;
#pragma unroll
      for (int i = 0; i < 8; ++i)
        hself[(gr0 + i) * 64 + c] = fmaxf(acc[t][i], 0.0f);
    }
  } else {
#pragma unroll
    for (int t = 0; t < 4; ++t) {
      int c = t * 16 + m;
#pragma unroll
      for (int i = 0; i < 8; ++i) {
        int gr = gr0 + i;
        if (gr < NN) hself[gr * 64 + c] = fmaxf(acc[t][i], 0.0f);
      }
    }
  }
}

// --- m = (efeat @ We + be) + h_self[src];  neigh[dst] += m  (atomic scatter) -
__global__ __launch_bounds__(256) void k_edge(
    const float* __restrict__ ef, const float* __restrict__ We,
    const float* __restrict__ be, const float* __restrict__ hself,
    const int* __restrict__ src, const int* __restrict__ dst,
    float* __restrict__ neigh) {
  __shared__ __align__(16) float sW[64 * 64];
  __shared__ __align__(16) float sA[128 * 68];  // 128 x 64, stride 68
  __shared__ int sSrc[128];
  __shared__ int sDst[128];
  const int tid = threadIdx.x;
  const int e0 = blockIdx.x * 128;

  // efeat tile: async global->LDS b128 (ASYNCcnt path), 128 rows x 16 float4
  for (int i = tid; i < 128 * 16; i += 256) {
    int r = i >> 4, q = i & 15;
    int e = e0 + r;
    if (e >= EE) e = EE - 1;
    async_g2l_b128(&ef[e * 64 + q * 4], &sA[r * 68 + q * 4]);
  }
  // weights + edge endpoints via regular vector loads
  for (int i = tid; i < 64 * 16; i += 256) {
    *reinterpret_cast<float4*>(&sW[i * 4]) =
        reinterpret_cast<const float4*>(We)[i];
  }
  if (tid < 128) {
    int e = e0 + tid;
    if (e >= EE) e = EE - 1;
    sSrc[tid] = src[e];
    sDst[tid] = dst[e];
  }
  wait_asynccnt0();
  __syncthreads();

  const int lane = tid & 31;
  const int wave = tid >> 5;
  const int m = lane & 15;
  const int koff = (lane >> 4) << 1;
  const int rsel = (lane >> 4) << 3;

  // prefetch the 16 gather rows this wave needs (hides HBM latency under WMMA)
  {
    int sn = sSrc[wave * 16 + m];
    const float* p = &hself[sn * 64];
    __builtin_prefetch(p, 0, 3);
    __builtin_prefetch(p + 32, 0, 3);
  }

  v8f acc[4];
#pragma unroll
  for (int t = 0; t < 4; ++t) {
    float b = be[t * 16 + m];
#pragma unroll
    for (int i = 0; i < 8; ++i) acc[t][i] = b;
  }

  const float* Arow = &sA[(wave * 16 + m) * 68];
#pragma unroll 4
  for (int k = 0; k < 64; k += 4) {
    v2f a;
    a.x = Arow[k + koff];
    a.y = Arow[k + koff + 1];
#pragma unroll
    for (int t = 0; t < 4; ++t) {
      v2f b;
      b.x = sW[(k + koff) * 64 + t * 16 + m];
      b.y = sW[(k + koff + 1) * 64 + t * 16 + m];
      acc[t] = wmma4(a, b, acc[t]);
    }
  }

  if (e0 + 128 <= EE) {  // uniform fast path (always for E = 9375*128)
#pragma unroll
    for (int t = 0; t < 4; ++t) {
      int c = t * 16 + m;
#pragma unroll
      for (int i = 0; i < 8; ++i) {
        int r = wave * 16 + rsel + i;
        float v = acc[t][i] + hself[sSrc[r] * 64 + c];
        atomicAdd(&neigh[sDst[r] * 64 + c], v);
      }
    }
  } else {
#pragma unroll
    for (int t = 0; t < 4; ++t) {
      int c = t * 16 + m;
#pragma unroll
      for (int i = 0; i < 8; ++i) {
        int r = wave * 16 + rsel + i;
        if (e0 + r < EE) {
          float v = acc[t][i] + hself[sSrc[r] * 64 + c];
          atomicAdd(&neigh[sDst[r] * 64 + c], v);
        }
      }
    }
  }
}

// -------- segmented cumsum over traversal order, scaled by 1/degree ---------
__global__ void k_run_init(int* __restrict__ rs) {
  int i = blockIdx.x * 256 + threadIdx.x;
  if (i < SS) rs[i] = NN;  // sentinel: segment empty
}
__global__ void k_run_fill(const int* __restrict__ seg, int* __restrict__ rs) {
  int i = blockIdx.x * 256 + threadIdx.x;
  if (i < NN) {
    int s = seg[i];
    if (i == 0 || seg[i - 1] != s) rs[s] = i;  // unique writer per segment
  }
}
__global__ __launch_bounds__(64) void k_segscan(
    const float* __restrict__ neigh, const int* __restrict__ perm,
    const int* __restrict__ seg, const float* __restrict__ degree,
    const int* __restrict__ rs, float* __restrict__ hneigh) {
  const int s = blockIdx.x;
  const int st = rs[s];
  if (st >= NN) return;
  const int c = threadIdx.x;  // one column per thread
  float run = 0.0f;
  for (int i = st; i < NN && seg[i] == s; ++i) {
    int node = perm[i];
    // prefetch next row of the chain while this one is in flight
    int nx = perm[(i + 1 < NN) ? (i + 1) : i];
    __builtin_prefetch(&neigh[nx * 64 + c], 0, 2);
    run += neigh[node * 64 + c];
    hneigh[node * 64 + c] = run / degree[node];
  }
}

// -------- x' = relu(h_self @ Ws + bs + h_neigh @ Wn + bn) -------------------
__global__ __launch_bounds__(128) void k_out(
    const float* __restrict__ hself, const float* __restrict__ hneigh,
    const float* __restrict__ Ws, const float* __restrict__ bs,
    const float* __restrict__ Wn, const float* __restrict__ bn,
    float* __restrict__ xout) {
  __shared__ __align__(16) float sA[64 * 68];
  __shared__ __align__(16) float sB[64 * 68];
  const int tid = threadIdx.x;
  const int row0 = blockIdx.x * 64;
  for (int i = tid; i < 64 * 16; i += 128) {
    int r = i >> 4, q = i & 15;
    int gr = row0 + r;
    if (gr >= NN) gr = NN - 1;
    *reinterpret_cast<float4*>(&sA[r * 68 + q * 4]) =
        reinterpret_cast<const float4*>(hself)[gr * 16 + q];
    *reinterpret_cast<float4*>(&sB[r * 68 + q * 4]) =
        reinterpret_cast<const float4*>(hneigh)[gr * 16 + q];
  }
  __syncthreads();

  const int lane = tid & 31;
  const int wave = tid >> 5;
  const int m = lane & 15;
  const int koff = (lane >> 4) << 1;
  const int rsel = (lane >> 4) << 3;

  v8f acc[4];
#pragma unroll
  for (int t = 0; t < 4; ++t) {
    float b = bs[t * 16 + m] + bn[t * 16 + m];
#pragma unroll
    for (int i = 0; i < 8; ++i) acc[t][i] = b;
  }

  const float* Ar = &sA[(wave * 16 + m) * 68];
  const float* Br = &sB[(wave * 16 + m) * 68];
#pragma unroll 4
  for (int k = 0; k < 64; k += 4) {
    v2f a;
    a.x = Ar[k + koff];
    a.y = Ar[k + koff + 1];
#pragma unroll
    for (int t = 0; t < 4; ++t) {
      v2f b;
      b.x = Ws[(k + koff) * 64 + t * 16 + m];
      b.y = Ws[(k + koff + 1) * 64 + t * 16 + m];
      acc[t] = wmma4(a, b, acc[t]);
    }
  }
#pragma unroll 4
  for (int k = 0; k < 64; k += 4) {
    v2f a;
    a.x = Br[k + koff];
    a.y = Br[k + koff + 1];
#pragma unroll
    for (int t = 0; t < 4; ++t) {
      v2f b;
      b.x = Wn[(k + koff) * 64 + t * 16 + m];
      b.y = Wn[(k + koff + 1) * 64 + t * 16 + m];
      acc[t] = wmma4(a, b, acc[t]);
    }
  }

  const int gr0 = row0 + wave * 16 + rsel;
  if (row0 + 64 <= NN) {
#pragma unroll
    for (int t = 0; t < 4; ++t) {
      int c = t * 16 + m;
#pragma unroll
      for (int i = 0; i < 8; ++i)
        xout[(gr0 + i) * 64 + c] = fmaxf(acc[t][i], 0.0f);
    }
  } else {
#pragma unroll
    for (int t = 0; t < 4; ++t) {
      int c = t * 16 + m;
#pragma unroll
      for (int i = 0; i < 8; ++i) {
        int gr = gr0 + i;
        if (gr < NN) xout[gr * 64 + c] = fmaxf(acc[t][i], 0.0f);
      }
    }
  }
}

__global__ void k_zero4(float4* __restrict__ p, int n4) {
  int i = blockIdx.x * 256 + threadIdx.x;
  if (i < n4) p[i] = make_float4(0.f, 0.f, 0.f, 0.f);
}

extern "C" void kernel_launch(void* const* d_in, const int* in_sizes, int n_in,
                              void* d_out, int out_size, void* d_ws,
                              size_t ws_size, hipStream_t stream) {
  const float* nfeat  = (const float*)d_in[0];
  const float* ts     = (const float*)d_in[1];
  const float* efeat  = (const float*)d_in[2];
  const float* degree = (const float*)d_in[3];
  const float* freq   = (const float*)d_in[4];
  const float* phase  = (const float*)d_in[5];
  const float* Wt     = (const float*)d_in[6];
  const float* bt     = (const float*)d_in[7];
  const float* We     = (const float*)d_in[8];
  const float* be     = (const float*)d_in[9];
  const float* Ws     = (const float*)d_in[10];
  const float* bs     = (const float*)d_in[11];
  const float* Wn     = (const float*)d_in[12];
  const float* bn     = (const float*)d_in[13];
  const int* src      = (const int*)d_in[14];
  const int* dst      = (const int*)d_in[15];
  const int* perm     = (const int*)d_in[16];
  const int* segid    = (const int*)d_in[17];
  float* out = (float*)d_out;

  // workspace layout
  float* hself  = (float*)d_ws;            // N*64
  float* neigh  = hself + NN * 64;         // N*64
  float* hneigh = neigh + NN * 64;         // N*64
  float* xbuf   = hneigh + NN * 64;        // N*64
  int* rs       = (int*)(xbuf + NN * 64);  // S

  // segment run starts (perm/segment_ids fixed across layers)
  k_run_init<<<(SS + 255) / 256, 256, 0, stream>>>(rs);
  k_run_fill<<<(NN + 255) / 256, 256, 0, stream>>>(segid, rs);

  for (int l = 0; l < 2; ++l) {
    const float* xin = (l == 0) ? nfeat : xbuf;
    float* xout = (l == 0) ? xbuf : out;

    k_zero4<<<(NN * 16 + 255) / 256, 256, 0, stream>>>((float4*)neigh, NN * 16);
    k_hself<<<(NN + 63) / 64, 128, 0, stream>>>(
        xin, ts, freq + l * 64, phase + l * 64, Wt + l * 128 * 64, bt + l * 64,
        hself);
    k_edge<<<(EE + 127) / 128, 256, 0, stream>>>(
        efeat, We + l * 64 * 64, be + l * 64, hself, src, dst, neigh);
    k_segscan<<<SS, 64, 0, stream>>>(neigh, perm, segid, degree, rs, hneigh);
    k_out<<<(NN + 63) / 64, 128, 0, stream>>>(
        hself, hneigh, Ws + l * 64 * 64, bs + l * 64, Wn + l * 64 * 64,
        bn + l * 64, xout);
  }
}